// SpatialTransformer2_6674379178142
// MI455X (gfx1250) — compile-verified
//
#include <hip/hip_runtime.h>

// Native clang vector so nontemporal builtins accept it (HIP float4 is a struct).
typedef float v4f __attribute__((ext_vector_type(4)));

// Problem dims fixed by the reference setup.
constexpr int Bn = 2, Cn = 2, Dn = 128, Hn = 224, Wn = 224;
constexpr int PLANE = Hn * Wn;          // 50176
constexpr int SLICE = Dn * PLANE;       // 6422528 (one [D,H,W] volume)
constexpr int QROW  = Wn / 4;           // 56 float4 quads per row
constexpr int ROWS_PER_BLOCK = 4;       // block = (56, 4) = 224 threads = 7 wave32s

__global__ __launch_bounds__(QROW * ROWS_PER_BLOCK) void warp_depth_lerp_kernel(
    const float* __restrict__ src,
    const float* __restrict__ flow,
    float* __restrict__ out)
{
    // Exact 3-D tiling: no div/rem, no bounds checks.
    const int xq = threadIdx.x;                                   // 0..55
    const int y  = blockIdx.x * ROWS_PER_BLOCK + threadIdx.y;     // 0..223
    const int z  = blockIdx.y;                                    // 0..127
    const int b  = blockIdx.z;                                    // 0..1

    const int x0     = xq << 2;              // multiple of 4 -> 16B aligned
    const int rowOff = y * Wn + x0;          // within a z-plane
    const int zOff   = z * PLANE + rowOff;   // within a [D,H,W] volume

    // Flow: streamed once; NT b128 load so it doesn't pollute L2 where src lives.
    const v4f f4 = __builtin_nontemporal_load(
        reinterpret_cast<const v4f*>(flow + b * SLICE + zOff));

    float wz[4];
    int   zi0[4], zi1[4];
    const float zf32 = (float)z;
#pragma unroll
    for (int i = 0; i < 4; ++i) {
        float zc = fminf(fmaxf(zf32 + f4[i], 0.0f), (float)(Dn - 1)); // border clamp
        float z0f = floorf(zc);
        wz[i]  = zc - z0f;
        zi0[i] = (int)z0f;
        zi1[i] = min(zi0[i] + 1, Dn - 1);
    }

    // Both channels share the flow quad: 2x reuse of the per-pixel weights.
#pragma unroll
    for (int c = 0; c < Cn; ++c) {
        const int volBase = (b * Cn + c) * SLICE;
        const float* __restrict__ sb = src + volBase + rowOff;
        v4f r;
#pragma unroll
        for (int i = 0; i < 4; ++i) {
            // Gather: per-lane z planes; src volume (103 MB) stays L2-resident (192 MB).
            float s0 = sb[zi0[i] * PLANE + i];
            float s1 = sb[zi1[i] * PLANE + i];
            r[i] = fmaf(wz[i], s1 - s0, s0);   // (1-w)*s0 + w*s1
        }
        // Output streamed once; NT b128 store keeps L2 for the src gather.
        __builtin_nontemporal_store(r,
            reinterpret_cast<v4f*>(out + volBase + zOff));
    }
}

extern "C" void kernel_launch(void* const* d_in, const int* in_sizes, int n_in,
                              void* d_out, int out_size, void* d_ws, size_t ws_size,
                              hipStream_t stream)
{
    const float* src  = (const float*)d_in[0];   // [B,C,D,H,W] f32
    const float* flow = (const float*)d_in[1];   // [B,1,D,H,W] f32
    float* out        = (float*)d_out;           // [B,C,D,H,W] f32

    dim3 block(QROW, ROWS_PER_BLOCK, 1);         // (56,4) = 224 threads
    dim3 grid(Hn / ROWS_PER_BLOCK, Dn, Bn);      // (56,128,2) exact
    warp_depth_lerp_kernel<<<grid, block, 0, stream>>>(src, flow, out);
}